// LossAF_36593121362214
// MI455X (gfx1250) — compile-verified
//
#include <hip/hip_runtime.h>
#include <hip/hip_bf16.h>
#include <cstdint>
#include <math.h>

#define NUM_CLASSES 80
#define DIMC (5 + NUM_CLASSES)
#define GMAX 32
#define TOPK 20
#define EPSF 1e-7f
#define AREA_MIN_F 3.2f     /* 4.0/1.25  */
#define AREA_MAX_F 320.0f   /* 256*1.25  */
#define IOU_W 3.0f
#define ACLS_W 0.5f
#define SIZE_W 0.2f
#define AR_W 0.1f
#define CEN_W 0.5f
#define CLS_SMOOTH 0.05f
#define PI_F 3.14159265358979323846f

typedef float v2f __attribute__((ext_vector_type(2)));
typedef float v8f __attribute__((ext_vector_type(8)));

__device__ __forceinline__ float sigmoidf_(float x) { return 1.0f / (1.0f + expf(-x)); }
__device__ __forceinline__ float softplusf_(float x) {
    return (x > 0.0f) ? (x + log1pf(expf(-x))) : log1pf(expf(x));
}
__device__ __forceinline__ float bcef_(float l, float t) {
    return softplusf_(-l) * t + softplusf_(l) * (1.0f - t);
}
__device__ __forceinline__ float ciou_(const float4 p, const float4 t) {
    float px1 = p.x, py1 = p.y, px2 = p.z, py2 = p.w;
    float tx1 = t.x, ty1 = t.y, tx2 = t.z, ty2 = t.w;
    float pw = fmaxf(px2 - px1, EPSF), ph = fmaxf(py2 - py1, EPSF);
    float tw = fmaxf(tx2 - tx1, EPSF), th = fmaxf(ty2 - ty1, EPSF);
    float iw = fmaxf(fminf(px2, tx2) - fmaxf(px1, tx1), 0.0f);
    float ih = fmaxf(fminf(py2, ty2) - fmaxf(py1, ty1), 0.0f);
    float inter = iw * ih;
    float uni = pw * ph + tw * th - inter + EPSF;
    float iou = inter / uni;
    float dx = px1 + px2 - (tx1 + tx2);
    float dy = py1 + py2 - (ty1 + ty2);
    float cd = dx * dx * 0.25f + dy * dy * 0.25f;
    float cw = fmaxf(px2, tx2) - fminf(px1, tx1);
    float ch = fmaxf(py2, ty2) - fminf(py1, ty1);
    float c2 = cw * cw + ch * ch + EPSF;
    float da = atanf(tw / th) - atanf(pw / ph);
    float v = (4.0f / (PI_F * PI_F)) * da * da;
    float alpha = v / (v - iou + 1.0f + EPSF);
    return iou - cd / c2 - alpha * v;
}

// ---------------------------------------------------------------------------
// Kernel 1: decode predicted boxes to xyxy (B128 store)
// ---------------------------------------------------------------------------
__global__ __launch_bounds__(256) void decode_kernel(
    const float* __restrict__ p, float4* __restrict__ predbox, int S, float stride)
{
    int b = blockIdx.y;
    int i = blockIdx.x * 256 + threadIdx.x;
    int Np = S * S;
    if (i >= Np) return;
    const float* pb = p + ((size_t)b * Np + i) * DIMC;
    float tx = pb[0], ty = pb[1], tw = pb[2], th = pb[3];
    float gx = (float)(i % S), gy = (float)(i / S);
    float px = (sigmoidf_(tx) * 2.0f - 0.5f + gx) * stride;
    float py = (sigmoidf_(ty) * 2.0f - 0.5f + gy) * stride;
    float pw = softplusf_(tw) * stride;
    float ph = softplusf_(th) * stride;
    predbox[(size_t)b * Np + i] =
        make_float4(px - pw * 0.5f, py - ph * 0.5f, px + pw * 0.5f, py + ph * 0.5f);
}

// ---------------------------------------------------------------------------
// Kernel 2: pair cost/IoU matrices. One wave per block; 16 preds x 32 gts.
// cdist^2 via two back-to-back V_WMMA_F32_16X16X4_F32; GT boxes staged into
// LDS with the CDNA5 async global->LDS path.
// ---------------------------------------------------------------------------
__global__ __launch_bounds__(32) void pair_kernel(
    const float* __restrict__ p, const float4* __restrict__ predbox,
    const float* __restrict__ gt_boxes, const int* __restrict__ gt_labels,
    const unsigned char* __restrict__ gt_mask,
    float* __restrict__ cost, float* __restrict__ iouc,
    int S, float stride)
{
    __shared__ float graw[GMAX * 4];
    __shared__ float g_x1[GMAX], g_y1[GMAX], g_x2[GMAX], g_y2[GMAX];
    __shared__ float g_cx[GMAX], g_cy[GMAX], g_w[GMAX], g_h[GMAX];
    __shared__ float g_c2[GMAX], g_gate[GMAX];
    __shared__ int   g_lab[GMAX];
    __shared__ float p_x1[16], p_y1[16], p_x2[16], p_y2[16];
    __shared__ float p_cx[16], p_cy[16], p_w[16], p_h[16], p_c2[16], p_so[16];

    const int b = blockIdx.y;
    const int pbase = blockIdx.x * 16;
    const int lane = threadIdx.x;
    const int Np = S * S;
    const float rad = 2.0f * stride;   // CENTER_RADIUS * stride

    // --- async copy of this image's 32 GT boxes (512B) into LDS: B128/lane ---
    {
        unsigned lds_off = (unsigned)(uintptr_t)(&graw[lane * 4]);
        unsigned long long ga =
            (unsigned long long)(uintptr_t)(gt_boxes + ((size_t)b * GMAX + lane) * 4);
        asm volatile("global_load_async_to_lds_b128 %0, %1, off"
                     :: "v"(lds_off), "v"(ga) : "memory");
    }

    // --- pred-tile stats (lanes 0..15) ---
    if (lane < 16) {
        int i = pbase + lane;
        float4 bx = predbox[(size_t)b * Np + i];
        float x1 = bx.x, y1 = bx.y, x2 = bx.z, y2 = bx.w;
        float cx = (x1 + x2) * 0.5f, cy = (y1 + y2) * 0.5f;
        float w = fmaxf(x2 - x1, EPSF), h = fmaxf(y2 - y1, EPSF);
        float ol = p[((size_t)b * Np + i) * DIMC + 4];
        p_x1[lane] = x1; p_y1[lane] = y1; p_x2[lane] = x2; p_y2[lane] = y2;
        p_cx[lane] = cx; p_cy[lane] = cy; p_w[lane] = w; p_h[lane] = h;
        p_c2[lane] = cx * cx + cy * cy;
        p_so[lane] = sigmoidf_(ol);
    }
    asm volatile("s_wait_asynccnt 0" ::: "memory");
    __syncthreads();

    // --- GT stats (all 32 lanes, one GT each) ---
    {
        int g = lane;
        float x1 = graw[g * 4 + 0], y1 = graw[g * 4 + 1];
        float x2 = graw[g * 4 + 2], y2 = graw[g * 4 + 3];
        float cx = (x1 + x2) * 0.5f, cy = (y1 + y2) * 0.5f;
        float w = fmaxf(x2 - x1, EPSF), h = fmaxf(y2 - y1, EPSF);
        float area = w * h / (stride * stride);
        unsigned char mk = gt_mask[b * GMAX + g];
        int lab = gt_labels[b * GMAX + g];
        lab = (lab < 0) ? 0 : ((lab > NUM_CLASSES - 1) ? NUM_CLASSES - 1 : lab);
        g_x1[g] = x1; g_y1[g] = y1; g_x2[g] = x2; g_y2[g] = y2;
        g_cx[g] = cx; g_cy[g] = cy; g_w[g] = w; g_h[g] = h;
        g_c2[g] = cx * cx + cy * cy;
        g_gate[g] = (area >= AREA_MIN_F && area <= AREA_MAX_F && mk != 0) ? 1.0f : 0.0f;
        g_lab[g] = lab;
    }
    __syncthreads();

    // --- A fragment: 16x4 f32, row m = [pcx, pcy, pcx^2+pcy^2, 1] ---
    const int mloc = lane & 15;
    v2f afrag;
    if (lane < 16) { afrag.x = p_cx[mloc]; afrag.y = p_cy[mloc]; }   // K=0,1
    else           { afrag.x = p_c2[mloc]; afrag.y = 1.0f;        }  // K=2,3

    #pragma unroll
    for (int t = 0; t < 2; ++t) {
        const int g = t * 16 + (lane & 15);
        // B fragment: 4x16 f32, col n = [-2gcx, -2gcy, 1, gcx^2+gcy^2]
        v2f bfrag;
        if (lane < 16) { bfrag.x = -2.0f * g_cx[g]; bfrag.y = 1.0f;     } // K=0,2
        else           { bfrag.x = -2.0f * g_cy[g]; bfrag.y = g_c2[g];  } // K=1,3
        v8f cacc = {};
        v8f dmat = __builtin_amdgcn_wmma_f32_16x16x4_f32(
            false, afrag, false, bfrag, (short)0, cacc, false, false);

        const float gx1 = g_x1[g], gy1 = g_y1[g], gx2 = g_x2[g], gy2 = g_y2[g];
        const float gcx = g_cx[g], gcy = g_cy[g], gw = g_w[g], gh = g_h[g];
        const float gate = g_gate[g];
        const int   lab = g_lab[g];
        const float ga2 = fmaxf(gx2 - gx1, 0.0f) * fmaxf(gy2 - gy1, 0.0f);
        const int   mhi = (lane < 16) ? 0 : 8;

        #pragma unroll
        for (int r = 0; r < 8; ++r) {
            const int m = r + mhi;
            const int i = pbase + m;
            float px1 = p_x1[m], py1 = p_y1[m], px2 = p_x2[m], py2 = p_y2[m];
            float pw = p_w[m], ph = p_h[m], so = p_so[m];
            // IoU
            float pa = fmaxf(px2 - px1, 0.0f) * fmaxf(py2 - py1, 0.0f);
            float iw = fmaxf(fminf(px2, gx2) - fmaxf(px1, gx1), 0.0f);
            float ih = fmaxf(fminf(py2, gy2) - fmaxf(py1, gy1), 0.0f);
            float inter = iw * ih;
            float iou = fminf(fmaxf(inter / (pa + ga2 - inter + EPSF), 0.0f), 1.0f);
            // candidacy: anchor center within radius, gate
            float ax = ((float)(i % S) + 0.5f) * stride;
            float ay = ((float)(i / S) + 0.5f) * stride;
            bool cand = (fabsf(ax - gcx) < rad) && (fabsf(ay - gcy) < rad) && (gate > 0.5f);
            // cls/obj cost
            float cl = p[((size_t)b * Np + i) * DIMC + 5 + lab];
            float cost_cls = -logf(sigmoidf_(cl) * so + EPSF);
            float size_cost = fabsf(logf(pw / gw)) + fabsf(logf(ph / gh));
            float ar_cost = fabsf(logf((pw / ph) * (gh / gw)));
            float cdist = sqrtf(fmaxf(dmat[r], 0.0f)) / stride;
            float c = IOU_W * (1.0f - iou) + ACLS_W * cost_cls + SIZE_W * size_cost
                    + AR_W * ar_cost + CEN_W * cdist + (cand ? 0.0f : 100000.0f);
            size_t idx = (((size_t)b * GMAX) + g) * Np + i;
            cost[idx] = c;
            iouc[idx] = cand ? iou : 0.0f;
        }
    }
}

// ---------------------------------------------------------------------------
// Kernel 3: per-GT dynamic-k and k-th smallest cost threshold (stable-sort
// exact via lexicographic (value, index) extraction).
// ---------------------------------------------------------------------------
__global__ __launch_bounds__(256) void pergt_kernel(
    const float* __restrict__ cost, const float* __restrict__ iouc,
    float* __restrict__ Tval, int* __restrict__ Tidx, int Np)
{
    const int g = blockIdx.x, b = blockIdx.y;
    const float* col_i = iouc + (((size_t)b * GMAX) + g) * Np;
    const float* col_c = cost + (((size_t)b * GMAX) + g) * Np;
    __shared__ float sv[256];
    __shared__ int si[256];
    const int tid = threadIdx.x;

    // Phase 1: sum of top-20 candidate IoUs -> k
    float prevV = INFINITY; int prevI = -1; float ssum = 0.0f;
    for (int it = 0; it < TOPK; ++it) {
        float bv = -INFINITY; int bi = 0x7fffffff;
        for (int i = tid; i < Np; i += 256) {
            float v = col_i[i];
            bool ok = (v < prevV) || (v == prevV && i > prevI);
            if (ok && (v > bv || (v == bv && i < bi))) { bv = v; bi = i; }
        }
        sv[tid] = bv; si[tid] = bi; __syncthreads();
        for (int s = 128; s > 0; s >>= 1) {
            if (tid < s) {
                float ov = sv[tid + s]; int oi = si[tid + s];
                if (ov > sv[tid] || (ov == sv[tid] && oi < si[tid])) { sv[tid] = ov; si[tid] = oi; }
            }
            __syncthreads();
        }
        prevV = sv[0]; prevI = si[0];
        ssum += prevV;
        __syncthreads();
    }
    int k = (int)ssum;
    k = (k < 1) ? 1 : ((k > TOPK) ? TOPK : k);

    // Phase 2: k-th smallest cost (with stable index tiebreak)
    prevV = -INFINITY; prevI = -1;
    for (int it = 0; it < k; ++it) {
        float bv = INFINITY; int bi = 0x7fffffff;
        for (int i = tid; i < Np; i += 256) {
            float v = col_c[i];
            bool ok = (v > prevV) || (v == prevV && i > prevI);
            if (ok && (v < bv || (v == bv && i < bi))) { bv = v; bi = i; }
        }
        sv[tid] = bv; si[tid] = bi; __syncthreads();
        for (int s = 128; s > 0; s >>= 1) {
            if (tid < s) {
                float ov = sv[tid + s]; int oi = si[tid + s];
                if (ov < sv[tid] || (ov == sv[tid] && oi < si[tid])) { sv[tid] = ov; si[tid] = oi; }
            }
            __syncthreads();
        }
        prevV = sv[0]; prevI = si[0];
        __syncthreads();
    }
    if (tid == 0) { Tval[b * GMAX + g] = prevV; Tidx[b * GMAX + g] = prevI; }
}

// ---------------------------------------------------------------------------
// Kernel 4: per-pred matching + losses -> deterministic block partials
// partial slot layout: [lbox, lcls, npos, lobj_scaled]
// ---------------------------------------------------------------------------
__global__ __launch_bounds__(256) void match_kernel(
    const float* __restrict__ p, const float4* __restrict__ predbox,
    const float4* __restrict__ gt_boxes, const int* __restrict__ gt_labels,
    const float* __restrict__ cost,
    const float* __restrict__ Tval, const int* __restrict__ Tidx,
    float* __restrict__ partials, int partial_base,
    int S, float inv_objsize)
{
    const int b = blockIdx.y;
    const int i = blockIdx.x * 256 + threadIdx.x;
    const int Np = S * S;
    const int tid = threadIdx.x;

    // stage per-GT thresholds once per block
    __shared__ float tV[GMAX];
    __shared__ int tI[GMAX];
    if (tid < GMAX) { tV[tid] = Tval[b * GMAX + tid]; tI[tid] = Tidx[b * GMAX + tid]; }
    __syncthreads();

    float lbox = 0.0f, lcls = 0.0f, npos = 0.0f, lobj = 0.0f;
    if (i < Np) {
        int nm = 0, firstg = -1, bestg = 0;
        float bestc = INFINITY;
        for (int g = 0; g < GMAX; ++g) {
            float c = cost[(((size_t)b * GMAX) + g) * Np + i];
            bool cand = c < 50000.0f;
            bool m = cand && (c < tV[g] || (c == tV[g] && i <= tI[g]));
            if (m) { nm++; if (firstg < 0) firstg = g; }
            if (c < bestc) { bestc = c; bestg = g; }
        }
        bool fg = nm > 0;
        int gidx = (nm > 1) ? bestg : (fg ? firstg : 0);
        float ol = p[((size_t)b * Np + i) * DIMC + 4];
        lobj = bcef_(ol, fg ? 1.0f : 0.0f) * inv_objsize;
        if (fg) {
            npos = 1.0f;
            float4 pb = predbox[(size_t)b * Np + i];
            float4 tb = gt_boxes[(size_t)b * GMAX + gidx];
            lbox = 1.0f - ciou_(pb, tb);
            int lab = gt_labels[b * GMAX + gidx];
            lab = (lab < 0) ? 0 : ((lab > NUM_CLASSES - 1) ? NUM_CLASSES - 1 : lab);
            const float* cl = p + ((size_t)b * Np + i) * DIMC + 5;
            const float off = CLS_SMOOTH / (float)(NUM_CLASSES - 1);
            for (int c = 0; c < NUM_CLASSES; ++c) {
                float tgt = off + ((c == lab) ? (1.0f - CLS_SMOOTH - off) : 0.0f);
                lcls += bcef_(cl[c], tgt);
            }
        }
    }

    __shared__ float s0[256], s1[256], s2[256], s3[256];
    s0[tid] = lbox; s1[tid] = lcls; s2[tid] = npos; s3[tid] = lobj;
    __syncthreads();
    for (int s = 128; s > 0; s >>= 1) {
        if (tid < s) {
            s0[tid] += s0[tid + s]; s1[tid] += s1[tid + s];
            s2[tid] += s2[tid + s]; s3[tid] += s3[tid + s];
        }
        __syncthreads();
    }
    if (tid == 0) {
        float* o = partials + (size_t)(partial_base + blockIdx.y * gridDim.x + blockIdx.x) * 4;
        o[0] = s0[0]; o[1] = s1[0]; o[2] = s2[0]; o[3] = s3[0];
    }
}

// ---------------------------------------------------------------------------
// Kernel 5: fixed-order final reduction -> scalar loss
// ---------------------------------------------------------------------------
__global__ __launch_bounds__(256) void final_kernel(
    const float* __restrict__ partials, int nslots, float* __restrict__ out)
{
    __shared__ float s0[256], s1[256], s2[256], s3[256];
    const int tid = threadIdx.x;
    float a = 0.0f, c = 0.0f, n = 0.0f, o = 0.0f;
    for (int i = tid; i < nslots; i += 256) {
        const float* q = partials + (size_t)i * 4;
        a += q[0]; c += q[1]; n += q[2]; o += q[3];
    }
    s0[tid] = a; s1[tid] = c; s2[tid] = n; s3[tid] = o;
    __syncthreads();
    for (int s = 128; s > 0; s >>= 1) {
        if (tid < s) {
            s0[tid] += s0[tid + s]; s1[tid] += s1[tid + s];
            s2[tid] += s2[tid + s]; s3[tid] += s3[tid + s];
        }
        __syncthreads();
    }
    if (tid == 0) {
        float denom = fmaxf(s2[0], 1.0f);
        out[0] = 5.0f * s0[0] / denom + 1.0f * s3[0] + 0.5f * s1[0] / denom;
    }
}

// ---------------------------------------------------------------------------
extern "C" void kernel_launch(void* const* d_in, const int* in_sizes, int n_in,
                              void* d_out, int out_size, void* d_ws, size_t ws_size,
                              hipStream_t stream) {
    (void)n_in; (void)out_size; (void)ws_size;
    const float* p3 = (const float*)d_in[0];
    const float* p4 = (const float*)d_in[1];
    const float* p5 = (const float*)d_in[2];
    const float* gtb = (const float*)d_in[3];
    const int* gtl = (const int*)d_in[4];
    const unsigned char* gtm = (const unsigned char*)d_in[5];
    float* out = (float*)d_out;

    const int Bn = in_sizes[3] / (GMAX * 4);   // batch from gt_boxes size
    const int SMAX = 80;
    const int NPMAX = SMAX * SMAX;

    char* ws = (char*)d_ws;
    size_t off = 0;
    auto take = [&](size_t bytes) { char* q = ws + off; off += (bytes + 255) & ~(size_t)255; return q; };

    // partial slots per level: ceil(Np/256) * Bn
    const int nb0 = (6400 + 255) / 256, nb1 = (1600 + 255) / 256, nb2 = (400 + 255) / 256;
    const int base0 = 0, base1 = nb0 * Bn, base2 = base1 + nb1 * Bn;
    const int nslots = base2 + nb2 * Bn;

    float* partials = (float*)take((size_t)nslots * 4 * sizeof(float));
    float4* predbox = (float4*)take((size_t)Bn * NPMAX * 4 * sizeof(float));
    float* cost     = (float*)take((size_t)Bn * GMAX * NPMAX * sizeof(float));
    float* iouc     = (float*)take((size_t)Bn * GMAX * NPMAX * sizeof(float));
    float* Tval     = (float*)take((size_t)Bn * GMAX * sizeof(float));
    int*   Tidx     = (int*)take((size_t)Bn * GMAX * sizeof(int));

    struct Lvl { const float* p; int S; float stride; int pbase; int nb; };
    const Lvl lv[3] = {
        {p3, 80,  8.0f, base0, nb0},
        {p4, 40, 16.0f, base1, nb1},
        {p5, 20, 32.0f, base2, nb2},
    };

    for (int l = 0; l < 3; ++l) {
        const int S = lv[l].S, Np = S * S;
        decode_kernel<<<dim3(lv[l].nb, Bn), 256, 0, stream>>>(lv[l].p, predbox, S, lv[l].stride);
        pair_kernel<<<dim3(Np / 16, Bn), 32, 0, stream>>>(
            lv[l].p, predbox, gtb, gtl, gtm, cost, iouc, S, lv[l].stride);
        pergt_kernel<<<dim3(GMAX, Bn), 256, 0, stream>>>(cost, iouc, Tval, Tidx, Np);
        match_kernel<<<dim3(lv[l].nb, Bn), 256, 0, stream>>>(
            lv[l].p, predbox, (const float4*)gtb, gtl, cost, Tval, Tidx,
            partials, lv[l].pbase, S, 1.0f / (float)((size_t)Bn * Np));
    }
    final_kernel<<<1, 256, 0, stream>>>(partials, nslots, out);
}